// Cheb_Conv_13649406067353
// MI455X (gfx1250) — compile-verified
//
#include <hip/hip_runtime.h>
#include <hip/hip_bf16.h>
#include <math.h>

#define N_NODES 50000
#define N_EDGES 800000
#define D_FEAT 128

typedef __attribute__((ext_vector_type(2))) float v2f;
typedef __attribute__((ext_vector_type(8))) float v8f;

// ---------------- utility kernels ----------------

__global__ void zero_kernel(float* __restrict__ p, int n) {
  int i = blockIdx.x * blockDim.x + threadIdx.x;
  if (i < n) p[i] = 0.f;
}

__global__ void deg_kernel(const int* __restrict__ ei, const float* __restrict__ ew,
                           float* __restrict__ deg) {
  int e = blockIdx.x * blockDim.x + threadIdx.x;
  if (e < N_EDGES) {
    int s = ei[e];  // src row of edge_index
    unsafeAtomicAdd(&deg[s], ew[e]);
  }
}

__global__ void dis_kernel(const float* __restrict__ deg, float* __restrict__ dis) {
  int i = blockIdx.x * blockDim.x + threadIdx.x;
  if (i < N_NODES) {
    float d = deg[i];
    dis[i] = d > 0.f ? rsqrtf(fmaxf(d, 1e-12f)) : 0.f;
  }
}

__global__ void what_kernel(const int* __restrict__ ei, const float* __restrict__ ew,
                            const float* __restrict__ dis, float* __restrict__ w_hat) {
  int e = blockIdx.x * blockDim.x + threadIdx.x;
  if (e < N_EDGES) {
    int s = ei[e];
    int d = ei[N_EDGES + e];
    w_hat[e] = -dis[s] * ew[e] * dis[d];
  }
}

// one wave per edge: 32 lanes x float4 = 128 features; coalesced b128 gather,
// hardware global_atomic_add_f32 scatter.
__global__ void prop_kernel(const int* __restrict__ ei, const float* __restrict__ w_hat,
                            const float* __restrict__ x, float* __restrict__ y) {
  long long idx = (long long)blockIdx.x * blockDim.x + threadIdx.x;
  long long total = (long long)N_EDGES * (D_FEAT / 4);
  if (idx >= total) return;
  int e = (int)(idx >> 5);          // D_FEAT/4 == 32 threads per edge
  int f = ((int)idx & 31) * 4;
  int s = ei[e];
  int d = ei[N_EDGES + e];
  float w = w_hat[e];
  const float4 v = *(const float4*)(x + (size_t)s * D_FEAT + f);
  float* yp = y + (size_t)d * D_FEAT + f;
  unsafeAtomicAdd(yp + 0, w * v.x);
  unsafeAtomicAdd(yp + 1, w * v.y);
  unsafeAtomicAdd(yp + 2, w * v.z);
  unsafeAtomicAdd(yp + 3, w * v.w);
}

// Weight prep: W0p = W0 - W2; W2p = 2*W2  (folds Tx2 = 2*P2 - x into the GEMM)
__global__ void wprep_kernel(const float* __restrict__ W0, const float* __restrict__ W2,
                             float* __restrict__ W0p, float* __restrict__ W2p, int n) {
  int i = blockIdx.x * blockDim.x + threadIdx.x;
  if (i < n) {
    float w2 = W2[i];
    W0p[i] = W0[i] - w2;
    W2p[i] = 2.f * w2;
  }
}

// ---------------- fused 3-term WMMA GEMM ----------------
// OUT[M,Cout] = leaky(X0@W0 + X1@W1 + X2@W2 + bias), fp32 via V_WMMA_F32_16X16X4_F32.
// blockDim = (32, Cout/16): each wave owns one 16x16 output tile; all waves share
// the same 16 A-rows (single A read per stripe). grid = (M/16, 1).
__global__ __launch_bounds__(256)
void cheb_gemm_wmma(const float* __restrict__ X0, const float* __restrict__ X1,
                    const float* __restrict__ X2,
                    const float* __restrict__ W0, const float* __restrict__ W1,
                    const float* __restrict__ W2,
                    const float* __restrict__ bias, float* __restrict__ OUT,
                    int Cin, int Cout) {
  const int lane = threadIdx.x;          // 0..31
  const bool hi  = lane >= 16;
  const int l15  = lane & 15;
  const int mBase = blockIdx.x * 16;
  const int nBase = threadIdx.y * 16;

  // A 16x4 f32 layout: lanes 0-15 hold (M=lane, K,K+1); lanes 16-31 hold (M=lane-16, K+2,K+3)
  const int aRow = mBase + l15;
  const int kOff = hi ? 2 : 0;
  // B 4x16 f32 layout: lane selects column N, VGPRs stripe K (mirror of A)
  const int nCol = nBase + l15;

  const float* As[3] = {X0, X1, X2};
  const float* Ws[3] = {W0, W1, W2};

  v8f acc = {};
  for (int t = 0; t < 3; ++t) {
    const float* A = As[t] + (size_t)aRow * Cin + kOff;
    const float* W = Ws[t] + (size_t)kOff * Cout + nCol;
#pragma unroll 8
    for (int k = 0; k < Cin; k += 4) {
      v2f a = *(const v2f*)(A + k);            // 8B contiguous, aligned
      v2f b;
      b.x = W[(size_t)k * Cout];
      b.y = W[(size_t)(k + 1) * Cout];
      acc = __builtin_amdgcn_wmma_f32_16x16x4_f32(false, a, false, b, (short)0, acc,
                                                  false, false);
    }
  }

  const float bv = bias[nCol];
  // C/D layout: VGPR j -> row mBase + j + (hi?8:0), col nCol
  float* outp = OUT + (size_t)(mBase + (hi ? 8 : 0)) * Cout + nCol;
#pragma unroll
  for (int j = 0; j < 8; ++j) {
    float v = acc[j] + bv;
    v = v > 0.f ? v : 0.01f * v;   // jax.nn.leaky_relu default slope
    outp[(size_t)j * Cout] = v;
  }
}

// ---------------- layer-3 GEMM fused with head ----------------
// blockDim = (32,4): 4 waves cover all 64 output cols of the same 16 rows.
// Per-lane contributions leaky(v)*Wl[col] are reduced in LDS (ds_add_f32),
// then sigmoid written directly — layer-3 output is never stored to global.
__global__ __launch_bounds__(128)
void cheb_gemm_wmma_head(const float* __restrict__ X0, const float* __restrict__ X1,
                         const float* __restrict__ X2,
                         const float* __restrict__ W0, const float* __restrict__ W1,
                         const float* __restrict__ W2,
                         const float* __restrict__ bias,
                         const float* __restrict__ Wl, const float* __restrict__ bl,
                         float* __restrict__ out, int Cin, int Cout) {
  __shared__ float hsum[16];
  const int tid = threadIdx.y * 32 + threadIdx.x;
  if (tid < 16) hsum[tid] = 0.f;
  __syncthreads();

  const int lane = threadIdx.x;
  const bool hi  = lane >= 16;
  const int l15  = lane & 15;
  const int mBase = blockIdx.x * 16;
  const int nBase = threadIdx.y * 16;

  const int aRow = mBase + l15;
  const int kOff = hi ? 2 : 0;
  const int nCol = nBase + l15;

  const float* As[3] = {X0, X1, X2};
  const float* Ws[3] = {W0, W1, W2};

  v8f acc = {};
  for (int t = 0; t < 3; ++t) {
    const float* A = As[t] + (size_t)aRow * Cin + kOff;
    const float* W = Ws[t] + (size_t)kOff * Cout + nCol;
#pragma unroll 8
    for (int k = 0; k < Cin; k += 4) {
      v2f a = *(const v2f*)(A + k);
      v2f b;
      b.x = W[(size_t)k * Cout];
      b.y = W[(size_t)(k + 1) * Cout];
      acc = __builtin_amdgcn_wmma_f32_16x16x4_f32(false, a, false, b, (short)0, acc,
                                                  false, false);
    }
  }

  const float bv = bias[nCol];
  const float wl = Wl[nCol];
  const int rOff = hi ? 8 : 0;
#pragma unroll
  for (int j = 0; j < 8; ++j) {
    float v = acc[j] + bv;
    v = v > 0.f ? v : 0.01f * v;        // leaky_relu on layer-3 output
    atomicAdd(&hsum[j + rOff], v * wl); // LDS ds_add_f32 reduction over 64 cols
  }
  __syncthreads();

  if (tid < 16) {
    float a = hsum[tid] + bl[0];
    out[mBase + tid] = 1.f / (1.f + expf(-a));
  }
}

// ---------------- launch ----------------

extern "C" void kernel_launch(void* const* d_in, const int* in_sizes, int n_in,
                              void* d_out, int out_size, void* d_ws, size_t ws_size,
                              hipStream_t stream) {
  const float* X    = (const float*)d_in[0];
  const int*   ei   = (const int*)d_in[1];     // edge_index [2,E]
  const float* ew   = (const float*)d_in[2];
  const float* W1_0 = (const float*)d_in[3];
  const float* W1_1 = (const float*)d_in[4];
  const float* W1_2 = (const float*)d_in[5];
  const float* b1   = (const float*)d_in[6];
  const float* W2_0 = (const float*)d_in[7];
  const float* W2_1 = (const float*)d_in[8];
  const float* W2_2 = (const float*)d_in[9];
  const float* b2   = (const float*)d_in[10];
  const float* W3_0 = (const float*)d_in[11];
  const float* W3_1 = (const float*)d_in[12];
  const float* W3_2 = (const float*)d_in[13];
  const float* b3   = (const float*)d_in[14];
  const float* Wl   = (const float*)d_in[15];
  const float* bl   = (const float*)d_in[16];
  float* out = (float*)d_out;

  float* ws = (float*)d_ws;
  float* deg   = ws; ws += N_NODES;
  float* dis   = ws; ws += N_NODES;
  float* w_hat = ws; ws += N_EDGES;
  float* T1    = ws; ws += (size_t)N_NODES * D_FEAT;
  float* T2    = ws; ws += (size_t)N_NODES * D_FEAT;
  float* Ha    = ws; ws += (size_t)N_NODES * D_FEAT;
  float* Hb    = ws; ws += (size_t)N_NODES * D_FEAT;
  float* W10p  = ws; ws += 128 * 128;
  float* W12p  = ws; ws += 128 * 128;
  float* W20p  = ws; ws += 128 * 128;
  float* W22p  = ws; ws += 128 * 128;
  float* W30p  = ws; ws += 128 * 64;
  float* W32p  = ws;

  const int TB = 256;
  const int nodeFeat = N_NODES * D_FEAT;
  const int gbNodeFeat = (nodeFeat + TB - 1) / TB;
  const int gbEdges = (N_EDGES + TB - 1) / TB;
  const int gbNodes = (N_NODES + TB - 1) / TB;
  const long long propThreads = (long long)N_EDGES * (D_FEAT / 4);
  const int gbProp = (int)((propThreads + TB - 1) / TB);

  // --- normalization: deg -> dis -> w_hat ---
  zero_kernel<<<gbNodes, TB, 0, stream>>>(deg, N_NODES);
  deg_kernel<<<gbEdges, TB, 0, stream>>>(ei, ew, deg);
  dis_kernel<<<gbNodes, TB, 0, stream>>>(deg, dis);
  what_kernel<<<gbEdges, TB, 0, stream>>>(ei, ew, dis, w_hat);

  // --- fold Tx2 = 2*P2 - x into the weights: W0p = W0 - W2, W2p = 2*W2 ---
  wprep_kernel<<<(128 * 128 + TB - 1) / TB, TB, 0, stream>>>(W1_0, W1_2, W10p, W12p, 128 * 128);
  wprep_kernel<<<(128 * 128 + TB - 1) / TB, TB, 0, stream>>>(W2_0, W2_2, W20p, W22p, 128 * 128);
  wprep_kernel<<<(128 * 64 + TB - 1) / TB, TB, 0, stream>>>(W3_0, W3_2, W30p, W32p, 128 * 64);

  auto prop = [&](const float* x, float* y) {
    zero_kernel<<<gbNodeFeat, TB, 0, stream>>>(y, nodeFeat);
    prop_kernel<<<gbProp, TB, 0, stream>>>(ei, w_hat, x, y);
  };

  dim3 blk128(32, 8);               // 8 waves: all 128 output cols per stripe
  dim3 blkHead(32, 4);              // 4 waves: all 64 output cols per stripe
  dim3 grd(N_NODES / 16, 1);        // 3125 row stripes exactly

  // --- layer 1: X -> Ha ---
  prop(X, T1);
  prop(T1, T2);                     // T2 = P2; Tx2 folded into weights
  cheb_gemm_wmma<<<grd, blk128, 0, stream>>>(X, T1, T2, W10p, W1_1, W12p, b1, Ha,
                                             128, 128);

  // --- layer 2: Ha -> Hb ---
  prop(Ha, T1);
  prop(T1, T2);
  cheb_gemm_wmma<<<grd, blk128, 0, stream>>>(Ha, T1, T2, W20p, W2_1, W22p, b2, Hb,
                                             128, 128);

  // --- layer 3 fused with head: Hb -> out (no N x 64 intermediate stored) ---
  prop(Hb, T1);
  prop(T1, T2);
  cheb_gemm_wmma_head<<<grd, blkHead, 0, stream>>>(Hb, T1, T2, W30p, W3_1, W32p, b3,
                                                   Wl, bl, out, 128, 64);
}